// Maxpooling_66700842107027
// MI455X (gfx1250) — compile-verified
//
#include <hip/hip_runtime.h>
#include <hip/hip_bf16.h>

// Problem constants (match reference)
#define BB 32
#define CC 512
#define TT 4096
#define OO 512

constexpr int THREADS = 256;   // 8 wave32s per block

// b128 async copy granule: 4 x i32, matching the builtin's expected pointee
typedef int v4i __attribute__((ext_vector_type(4)));
typedef __attribute__((address_space(1))) v4i* gvec_ptr;   // global
typedef __attribute__((address_space(3))) v4i* lvec_ptr;   // LDS

// One block per (b, c) row. Bandwidth-bound: stage the row in LDS via the
// CDNA5 async global->LDS path, then do the windowed max with aligned
// ds_load_b128 reads (window [s,s+9) always fits in 3 aligned float4 blocks).
__global__ __launch_bounds__(THREADS)
void Maxpooling_66700842107027_kernel(const float* __restrict__ x,
                                      const int* __restrict__ length,
                                      float* __restrict__ out) {
    __shared__ __align__(16) float lds[TT];   // 16 KB row buffer

    const int row = blockIdx.x;               // row = b*CC + c
    const int b   = row / CC;
    const int tid = threadIdx.x;
    const int L   = length[b];                // 1 <= L <= TT-1

    const float* __restrict__ xrow = x   + (size_t)row * TT;
    float* __restrict__       orow = out + (size_t)row * OO;

    if (L >= OO) {
        // ---- pool branch: stage x[b,c,0:round4(L)] into LDS ----
        const int nvec = (L + 3) >> 2;        // # of 16B chunks (<= 1024)
#if __has_builtin(__builtin_amdgcn_global_load_async_to_lds_b128) && \
    __has_builtin(__builtin_amdgcn_s_wait_asynccnt)
        for (int i = tid; i < nvec; i += THREADS) {
            __builtin_amdgcn_global_load_async_to_lds_b128(
                (gvec_ptr)(xrow + i * 4),
                (lvec_ptr)(lds + i * 4),
                /*offset=*/0, /*cpol=NT*/1);
        }
        __builtin_amdgcn_s_wait_asynccnt(0);  // my async stores to LDS done
        __syncthreads();                       // everyone's are done
#else
        for (int i = tid; i < nvec; i += THREADS) {
            *(float4*)(lds + i * 4) = *(const float4*)(xrow + i * 4);
        }
        __syncthreads();
#endif

        const float4* lds4 = (const float4*)lds;

        // ---- windowed max: 2 outputs per thread ----
        #pragma unroll
        for (int jj = 0; jj < OO / THREADS; ++jj) {
            const int j = tid + jj * THREADS;
            // start = floor(j*L/O), end = ceil((j+1)*L/O); O=512 -> shifts
            const unsigned s = (unsigned)(j * L) / OO;
            const unsigned e = (unsigned)((j + 1) * L + (OO - 1)) / OO;
            const unsigned a = s & ~3u;       // aligned base; a+11 <= 4095
            // three aligned b128 LDS loads cover [a, a+12) >= [s, s+9)
            const float4 v0 = lds4[(a >> 2) + 0];
            const float4 v1 = lds4[(a >> 2) + 1];
            const float4 v2 = lds4[(a >> 2) + 2];
            float w[12] = { v0.x, v0.y, v0.z, v0.w,
                            v1.x, v1.y, v1.z, v1.w,
                            v2.x, v2.y, v2.z, v2.w };
            float m = -__builtin_inff();
            #pragma unroll
            for (int k = 0; k < 12; ++k) {
                const unsigned p = a + (unsigned)k;
                const bool valid = (p >= s) & (p < e);
                m = valid ? fmaxf(m, w[k]) : m;   // mask outside-bin lanes
            }
            orow[j] = m;
        }
    } else {
        // ---- copy branch: out[j] = j < L ? x[j] : 0, vectorized b128 ----
        if (tid < OO / 4) {
            const int j0 = tid * 4;
            float4 v = *(const float4*)(xrow + j0);   // j0+3 <= 511 < TT
            v.x = (j0 + 0 < L) ? v.x : 0.0f;
            v.y = (j0 + 1 < L) ? v.y : 0.0f;
            v.z = (j0 + 2 < L) ? v.z : 0.0f;
            v.w = (j0 + 3 < L) ? v.w : 0.0f;
            *(float4*)(orow + j0) = v;
        }
    }
}

extern "C" void kernel_launch(void* const* d_in, const int* in_sizes, int n_in,
                              void* d_out, int out_size, void* d_ws, size_t ws_size,
                              hipStream_t stream) {
    const float* x      = (const float*)d_in[0];   // [B, C, T] f32
    const int*   length = (const int*)d_in[1];     // [B] i32
    float*       out    = (float*)d_out;           // [B, C, O] f32
    (void)in_sizes; (void)n_in; (void)out_size; (void)d_ws; (void)ws_size;

    dim3 grid(BB * CC);     // one block per (b, c) row
    dim3 block(THREADS);
    Maxpooling_66700842107027_kernel<<<grid, block, 0, stream>>>(x, length, out);
}